// JPEGCodecDifferentiable_61684320305794
// MI455X (gfx1250) — compile-verified
//
#include <hip/hip_runtime.h>
#include <math.h>

typedef float v2f __attribute__((ext_vector_type(2)));
typedef float v8f __attribute__((ext_vector_type(8)));

#define PI_OVER_16 0.19634954084936207f

__device__ __forceinline__ float dpoly_round(float x){ float r = rintf(x);  float d = x - r;        return r + d*d*d; }
__device__ __forceinline__ float dpoly_floor(float x){ float f = floorf(x); float d = x - 0.5f - f; return f + d*d*d; }
__device__ __forceinline__ float ddiff_clip(float x){
  x = (x > 255.f) ? (-0.02f*(__expf(255.f - x) - 1.f) + 255.f) : x;
  x = (x <   1.f) ? ( 0.02f*(__expf(x - 1.f)   - 1.f) +   1.f) : x;
  return x;
}
__device__ __forceinline__ float scaled_qt_entry(float qt, float q){
  float sc = dpoly_floor((q < 50.f) ? (5000.f / q) : (200.f - 2.f*q));
  return dpoly_floor(ddiff_clip((qt*sc + 50.f) / 100.f));
}

// ---------------------------------------------------------------------------
// K1: RGB -> YCbCr (x255). Y full-res into out ch0; Cb/Cr AA-downsampled 2x.
// ---------------------------------------------------------------------------
__global__ void __launch_bounds__(256)
k_color_down(const float* __restrict__ rgb, float* __restrict__ yout,
             float* __restrict__ cb_ds, float* __restrict__ cr_ds,
             int B, int H, int W)
{
  const int Hh = H >> 1, Wh = W >> 1;
  int idx = blockIdx.x * 256 + threadIdx.x;
  int total = B * Hh * Wh;
  if (idx >= total) return;
  int j = idx % Wh; int t = idx / Wh; int i = t % Hh; int b = t / Hh;

  const size_t HW = (size_t)H * W;
  const float* rp = rgb + (size_t)b * 3 * HW;
  const float* gp = rp + HW;
  const float* bp = rp + 2 * HW;

  const float wts[4] = {1.f, 3.f, 3.f, 1.f};
  float cbacc = 0.f, cracc = 0.f;
  float yv[2][2];
  for (int dy = 0; dy < 4; ++dy){
    int ry = 2*i - 1 + dy; ry = ry < 0 ? 0 : (ry > H-1 ? H-1 : ry);
    float rowcb = 0.f, rowcr = 0.f;
    for (int dx = 0; dx < 4; ++dx){
      int rx = 2*j - 1 + dx; rx = rx < 0 ? 0 : (rx > W-1 ? W-1 : rx);
      size_t o = (size_t)ry * W + rx;
      float r = rp[o], g = gp[o], bb = bp[o];
      float yy = 0.299f*r + 0.587f*g + 0.114f*bb;
      float cb = (bb - yy)*0.564f + 0.5f;
      float cr = (r  - yy)*0.713f + 0.5f;
      rowcb += wts[dx]*cb; rowcr += wts[dx]*cr;
      if (dy >= 1 && dy <= 2 && dx >= 1 && dx <= 2) yv[dy-1][dx-1] = 255.f * yy;
    }
    cbacc += wts[dy]*rowcb; cracc += wts[dy]*rowcr;
  }
  float* yb = yout + (size_t)b * 3 * HW;   // channel 0 of output buffer
  yb[(size_t)(2*i  )*W + 2*j    ] = yv[0][0];
  yb[(size_t)(2*i  )*W + 2*j + 1] = yv[0][1];
  yb[(size_t)(2*i+1)*W + 2*j    ] = yv[1][0];
  yb[(size_t)(2*i+1)*W + 2*j + 1] = yv[1][1];
  size_t co = (size_t)b * Hh * Wh + (size_t)i * Wh + j;
  cb_ds[co] = 255.f * cbacc * (1.f/64.f);
  cr_ds[co] = 255.f * cracc * (1.f/64.f);
}

// ---------------------------------------------------------------------------
// K2/K3: per-wave 16x16 tile (= 2x2 JPEG 8x8 blocks) DCT -> quant -> IDCT,
// in place, using chained V_WMMA_F32_16X16X4_F32 with G = blockdiag(C8, C8).
// ---------------------------------------------------------------------------
__global__ void __launch_bounds__(256)
k_jpeg_blocks(float* __restrict__ plane, size_t batchStride,
              const float* __restrict__ qt, const float* __restrict__ quality,
              int Wp, int tx, int ty, int B)
{
  __shared__ __align__(16) float lds_all[8 * 256];
  const int lane = threadIdx.x & 31;
  const int wave = threadIdx.x >> 5;
  float* wl = lds_all + wave * 256;

  int gw = blockIdx.x * (blockDim.x >> 5) + wave;
  int numTiles = B * tx * ty;
  if (gw >= numTiles) return;                 // wave-uniform: EXEC stays all-ones
  int b   = gw / (tx * ty);
  int rem = gw - b * tx * ty;
  int tyi = rem / tx;
  int txi = rem - tyi * tx;
  float* base = plane + (size_t)b * batchStride + (size_t)tyi * 16 * Wp + (size_t)txi * 16;

  const int col  = lane & 15;        // column index (B/C/D layouts)
  const int rsel = (lane >> 4) << 1; // K sub-select 0 or 2 (A/B layouts)
  const int hi8  = (lane >> 4) << 3; // row offset 0 or 8 (C/D layout)

  // ---- load X - 128 in B-layout: bm[s][t] = X[4s+rsel+t][col] - 128 -------
  float bm[4][2];
  for (int s = 0; s < 4; ++s){
    int r0 = 4*s + rsel;
    bm[s][0] = base[(size_t)r0     * Wp + col] - 128.f;
    bm[s][1] = base[(size_t)(r0+1) * Wp + col] - 128.f;
  }

  // ---- constant A operands: Gt = G^T (DCT), G (IDCT), per A-layout --------
  float adct[4][2], aidt[4][2];
  for (int s = 0; s < 4; ++s)
    for (int t = 0; t < 2; ++t){
      int k = 4*s + rsel + t;
      float cd = __cosf((2.f*(float)(k   & 7) + 1.f) * (float)(col & 7) * PI_OVER_16); // C8[k&7][m&7]
      float ci = __cosf((2.f*(float)(col & 7) + 1.f) * (float)(k  & 7) * PI_OVER_16); // C8[m&7][k&7]
      bool same = ((k >> 3) == (col >> 3));
      adct[s][t] = same ? cd : 0.f;
      aidt[s][t] = same ? ci : 0.f;
    }

  v8f acc;
  // acc = A * bm  (16x16x16 via four chained 16x16x4 fp32 WMMAs)
#define MM16(AARR) do {                                                         \
    acc = (v8f){0.f,0.f,0.f,0.f,0.f,0.f,0.f,0.f};                               \
    for (int s_ = 0; s_ < 4; ++s_){                                             \
      v2f av_ = { AARR[s_][0], AARR[s_][1] };                                   \
      v2f bv_ = { bm[s_][0],  bm[s_][1]  };                                     \
      acc = __builtin_amdgcn_wmma_f32_16x16x4_f32(false, av_, false, bv_,       \
                                                  (short)0, acc, false, false); \
    } } while (0)

  // D-layout(acc = H) -> B-layout of H^T, via LDS (transpose fold-in)
#define DTOB() do {                                                             \
    asm volatile("" ::: "memory");                                              \
    for (int j_ = 0; j_ < 8; ++j_) wl[(j_ + hi8) * 16 + col] = acc[j_];         \
    asm volatile("s_wait_dscnt 0x0" ::: "memory");                              \
    for (int s_ = 0; s_ < 4; ++s_){                                             \
      int k0_ = 4*s_ + rsel;                                                    \
      bm[s_][0] = wl[col * 16 + k0_    ];                                       \
      bm[s_][1] = wl[col * 16 + k0_ + 1];                                       \
    }                                                                           \
    asm volatile("s_wait_dscnt 0x0" ::: "memory");                              \
  } while (0)

  MM16(adct);          // T = G^T * (X-128)               (D-layout: T)
  DTOB();              // bm = B-layout of T^T
  MM16(adct);          // W = G^T * T^T = (T*G)^T = U^T   (D-layout: U^T)

  // ---- quant / soft-round / dequant (elementwise on U^T) ------------------
  {
    float q = quality[b];
    int u8 = lane & 7;                              // DCT coeff row u
    float alpha_u = (u8 == 0) ? 0.70710678118654752f : 1.f;
    for (int j = 0; j < 8; ++j){
      int v8 = j;                                   // DCT coeff col v
      float alpha_v = (v8 == 0) ? 0.70710678118654752f : 1.f;
      float sv = 0.25f * alpha_u * alpha_v;         // DCT 0.25*S and IDCT 0.25*S
      float qv = scaled_qt_entry(qt[u8 * 8 + v8], q);
      float u_true = sv * acc[j];
      float e = dpoly_round(u_true / qv);
      acc[j] = sv * (e * qv);                       // Mid^T element
    }
  }

  DTOB();              // bm = B-layout of Mid
  MM16(aidt);          // V1 = G * Mid                    (D-layout: V1)
  DTOB();              // bm = B-layout of V1^T
  MM16(aidt);          // G * V1^T = (V1*G^T)^T = R^T     (D-layout: R^T)

  // ---- +128, transpose into LDS row-major, coalesced float4 store ---------
  asm volatile("" ::: "memory");
  for (int j = 0; j < 8; ++j)
    wl[col * 16 + (j + hi8)] = acc[j] + 128.f;       // pixel(row=col, col=j+hi8)
  asm volatile("s_wait_dscnt 0x0" ::: "memory");

  int rr = lane >> 1;
  int hc = (lane & 1) * 8;
  float4 f0 = *(const float4*)&wl[rr * 16 + hc];
  float4 f1 = *(const float4*)&wl[rr * 16 + hc + 4];
  *(float4*)&base[(size_t)rr * Wp + hc    ] = f0;
  *(float4*)&base[(size_t)rr * Wp + hc + 4] = f1;
#undef MM16
#undef DTOB
}

// ---------------------------------------------------------------------------
// K4: chroma 2x upsample + YCbCr -> RGB + clip. Y read from out ch0 (own px).
// ---------------------------------------------------------------------------
__global__ void __launch_bounds__(256)
k_up_color(float* __restrict__ out, const float* __restrict__ cb_ds,
           const float* __restrict__ cr_ds, int B, int H, int W)
{
  int idx = blockIdx.x * 256 + threadIdx.x;
  int total = B * H * W;
  if (idx >= total) return;
  int x = idx % W; int t = idx / W; int y = t % H; int b = t / H;
  const int Hh = H >> 1, Wh = W >> 1;

  int iy = y >> 1, ix = x >> 1;
  int y0, y1, x0, x1; float wy0, wy1, wx0, wx1;
  if ((y & 1) == 0){ y0 = (iy-1 < 0) ? 0 : iy-1; y1 = iy;                         wy0 = 0.25f; wy1 = 0.75f; }
  else             { y0 = iy;                    y1 = (iy+1 > Hh-1) ? Hh-1 : iy+1; wy0 = 0.75f; wy1 = 0.25f; }
  if ((x & 1) == 0){ x0 = (ix-1 < 0) ? 0 : ix-1; x1 = ix;                         wx0 = 0.25f; wx1 = 0.75f; }
  else             { x0 = ix;                    x1 = (ix+1 > Wh-1) ? Wh-1 : ix+1; wx0 = 0.75f; wx1 = 0.25f; }

  const float* cbb = cb_ds + (size_t)b * Hh * Wh;
  const float* crb = cr_ds + (size_t)b * Hh * Wh;
  float cb = wy0 * (wx0 * cbb[(size_t)y0*Wh + x0] + wx1 * cbb[(size_t)y0*Wh + x1])
           + wy1 * (wx0 * cbb[(size_t)y1*Wh + x0] + wx1 * cbb[(size_t)y1*Wh + x1]);
  float cr = wy0 * (wx0 * crb[(size_t)y0*Wh + x0] + wx1 * crb[(size_t)y0*Wh + x1])
           + wy1 * (wx0 * crb[(size_t)y1*Wh + x0] + wx1 * crb[(size_t)y1*Wh + x1]);

  const size_t HW = (size_t)H * W;
  float* ob = out + (size_t)b * 3 * HW;
  size_t po = (size_t)y * W + x;
  float yn  = ob[po] * (1.f/255.f);
  float cbn = cb * (1.f/255.f) - 0.5f;
  float crn = cr * (1.f/255.f) - 0.5f;
  float r = yn + 1.403f * crn;
  float g = yn - 0.714f * crn - 0.344f * cbn;
  float bb = yn + 1.773f * cbn;
  r  = fminf(fmaxf(r , 0.f), 1.f);
  g  = fminf(fmaxf(g , 0.f), 1.f);
  bb = fminf(fmaxf(bb, 0.f), 1.f);
  ob[po]          = r;
  ob[HW + po]     = g;
  ob[2*HW + po]   = bb;
}

// ---------------------------------------------------------------------------
extern "C" void kernel_launch(void* const* d_in, const int* in_sizes, int n_in,
                              void* d_out, int out_size, void* d_ws, size_t ws_size,
                              hipStream_t stream)
{
  const float* rgb     = (const float*)d_in[0];
  const float* quality = (const float*)d_in[1];
  const float* qt_y    = (const float*)d_in[2];
  const float* qt_c    = (const float*)d_in[3];
  float* out = (float*)d_out;

  const int B = 32, H = 512, W = 512;
  const int Hh = H / 2, Wh = W / 2;

  float* cb = (float*)d_ws;                         // B*Hh*Wh floats (8 MB)
  float* cr = cb + (size_t)B * Hh * Wh;             // 8 MB more

  // K1: color convert + chroma downsample (Y -> out channel 0)
  int t1 = B * Hh * Wh;
  k_color_down<<<(t1 + 255) / 256, 256, 0, stream>>>(rgb, out, cb, cr, B, H, W);

  // K2: luma DCT/quant/IDCT in place (plane = out ch0, batch stride 3*H*W)
  int txY = W / 16, tyY = H / 16;
  int ntY = B * txY * tyY;                           // 32768 waves
  k_jpeg_blocks<<<(ntY * 32 + 255) / 256, 256, 0, stream>>>(
      out, (size_t)3 * H * W, qt_y, quality, W, txY, tyY, B);

  // K3: chroma DCT/quant/IDCT in place (two planes)
  int txC = Wh / 16, tyC = Hh / 16;
  int ntC = B * txC * tyC;                           // 8192 waves each
  k_jpeg_blocks<<<(ntC * 32 + 255) / 256, 256, 0, stream>>>(
      cb, (size_t)Hh * Wh, qt_c, quality, Wh, txC, tyC, B);
  k_jpeg_blocks<<<(ntC * 32 + 255) / 256, 256, 0, stream>>>(
      cr, (size_t)Hh * Wh, qt_c, quality, Wh, txC, tyC, B);

  // K4: upsample + color + clip
  int t4 = B * H * W;
  k_up_color<<<(t4 + 255) / 256, 256, 0, stream>>>(out, cb, cr, B, H, W);
}